// QuantumSelfAttention_65481071408163
// MI455X (gfx1250) — compile-verified
//
#include <hip/hip_runtime.h>
#include <math.h>

// Problem constants (from reference)
#define BB  8
#define SS  2048
#define EE  1024
#define NWQ 8

typedef __attribute__((ext_vector_type(2))) float v2f;
typedef __attribute__((ext_vector_type(8))) float v8f;

// D = A(16x4 f32) * B(4x16 f32) + C(16x16 f32), exact fp32 matrix op.
__device__ __forceinline__ v8f wmma_f32_4(v2f a, v2f b, v8f c) {
  return __builtin_amdgcn_wmma_f32_16x16x4_f32(
      /*neg_a=*/false, a, /*neg_b=*/false, b,
      /*c_mod=*/(short)0, c, /*reuse_a=*/false, /*reuse_b=*/false);
}

// Async global -> LDS copy of 16 bytes per lane (CDNA5 GLOBAL_LOAD_ASYNC_TO_LDS_B128,
// tracked by ASYNCcnt; LDS written directly, no VGPR bounce).
__device__ __forceinline__ void async_copy_b128(unsigned lds_addr, const void* gptr) {
  asm volatile("global_load_async_to_lds_b128 %0, %1, off"
               :: "v"(lds_addr), "v"((unsigned long long)(uintptr_t)gptr)
               : "memory");
}

// -------------------------------------------------------------------------
// Kernel 1: qq/kq/vq = cos(x @ W^T + b + theta) for W in {wq,wk,wv}
// One pass over x (64 MB) = the HBM-bound leg. 64 rows per workgroup,
// 8 waves = 4 row-tiles x 2 N-tiles (N padded 24 -> 32).
// Double-buffered LDS, filled by async global->LDS B128 copies so the
// next K-chunk streams in underneath the WMMA loop of the current one.
// Stride-68 row padding -> conflict-free ds_load_b64 fragment reads.
// -------------------------------------------------------------------------
__global__ __launch_bounds__(256) void qkv_proj_kernel(
    const float* __restrict__ x,
    const float* __restrict__ wq, const float* __restrict__ bq,
    const float* __restrict__ wk, const float* __restrict__ bk,
    const float* __restrict__ wv, const float* __restrict__ bv,
    const float* __restrict__ theta,
    float* __restrict__ qq, float* __restrict__ kq, float* __restrict__ vq)
{
  __shared__ float xs[2][64][68];   // 2 x (64 rows x 64-col K-chunk, +pad)
  __shared__ float ws[2][32][68];   // 2 x (24 weight rows + 8 zero rows)

  const int tid     = threadIdx.x;
  const int lane    = tid & 31;
  const int wave    = tid >> 5;
  const int m       = lane & 15;   // row (A) / col (B,D) within 16x16 tile
  const int h       = lane >> 4;   // K-half selector
  const int rowTile = wave & 3;
  const int nTile   = wave >> 2;
  const int r0      = blockIdx.x * 64;

  // Zero the 8 dead weight rows (24..31) of both buffers once, via DS stores.
  // Async staging never touches these rows.
  for (int i = tid; i < 2 * 8 * 64; i += 256) {
    int bufi = i >> 9;
    int r    = 24 + ((i >> 6) & 7);
    int cc   = i & 63;
    ws[bufi][r][cc] = 0.0f;
  }

  // Issue one K-chunk's staging as async copies: 4 (x) + 2 (w) = 6 async
  // instructions per wave, uniformly (dead weight slots remap to rows 0..7,
  // duplicating identical bytes to identical LDS addresses -- benign).
  auto issue_stage = [&](int buf, int kc) {
    #pragma unroll
    for (int it = 0; it < 4; ++it) {
      int i   = tid + it * 256;
      int row = i >> 4;
      int cg  = (i & 15) << 2;
      async_copy_b128((unsigned)(uintptr_t)&xs[buf][row][cg],
                      x + (size_t)(r0 + row) * EE + kc + cg);
    }
    #pragma unroll
    for (int it = 0; it < 2; ++it) {
      int i   = tid + it * 256;
      int j   = (i < 384) ? i : (i - 384);   // remap dead slots -> rows 0..7
      int row = j >> 4;                      // 0..23
      int cg  = (j & 15) << 2;
      const float* gp;
      if (row < 8)       gp = wq + (size_t)row        * EE + kc + cg;
      else if (row < 16) gp = wk + (size_t)(row - 8)  * EE + kc + cg;
      else               gp = wv + (size_t)(row - 16) * EE + kc + cg;
      async_copy_b128((unsigned)(uintptr_t)&ws[buf][row][cg], gp);
    }
  };

  v8f acc = {};

  issue_stage(0, 0);                         // prologue: chunk 0 -> buf 0

  for (int c = 0; c < EE / 64; ++c) {
    const int buf = c & 1;
    if (c + 1 < EE / 64) {
      issue_stage(buf ^ 1, (c + 1) * 64);    // stream next chunk under compute
      asm volatile("s_wait_asynccnt 0x6" ::: "memory");  // current chunk landed
    } else {
      asm volatile("s_wait_asynccnt 0x0" ::: "memory");
    }
    __syncthreads();                         // all waves' copies visible

    // 16 fp32 WMMAs per K-chunk per wave.
    #pragma unroll
    for (int kk = 0; kk < 64; kk += 4) {
      v2f a, b;
      const float* ap = &xs[buf][rowTile * 16 + m][kk + 2 * h];
      const float* bp = &ws[buf][nTile   * 16 + m][kk + 2 * h];
      a.x = ap[0]; a.y = ap[1];      // A[m][2h], A[m][2h+1]
      b.x = bp[0]; b.y = bp[1];      // B[2h][m], B[2h+1][m] (= Wcat[m][k])
      acc = wmma_f32_4(a, b, acc);
    }
    __syncthreads();                         // free this buffer for refill
  }

  // Epilogue: bias + theta + cos, scatter into qq/kq/vq [B*S, 8].
  const int col = nTile * 16 + m;            // D-layout: lane = column
  if (col < 24) {
    const int c8 = col & 7;
    float bias;
    float* dst;
    if (col < 8)       { bias = bq[c8]; dst = qq; }
    else if (col < 16) { bias = bk[c8]; dst = kq; }
    else               { bias = bv[c8]; dst = vq; }
    const float add = bias + theta[c8];
    #pragma unroll
    for (int j = 0; j < 8; ++j) {
      int row = r0 + rowTile * 16 + j + 8 * h;   // D-layout: VGPR j -> rows j/j+8
      dst[(size_t)row * NWQ + c8] = __cosf(acc[j] + add);
    }
  }
}

// -------------------------------------------------------------------------
// Kernel 2: fused flash-attention + output projection.
// grid = (S/128, B); 8 waves/WG; each wave owns 16 query rows.
// Per 16-key tile: 2 WMMAs (scores, K=8), online softmax via __shfl_xor
// butterflies inside each 16-lane half, LDS D->A transpose (s_wait_dscnt),
// 4 WMMAs (P@V, K=16). Final: (O/L) @ wc^T + bc, 2 WMMAs x 64 col tiles,
// coalesced fp32 stores. Never materializes [B,S,S].
// -------------------------------------------------------------------------
__global__ __launch_bounds__(256) void attn_out_kernel(
    const float* __restrict__ qq, const float* __restrict__ kq,
    const float* __restrict__ vq,
    const float* __restrict__ wc, const float* __restrict__ bc,
    float* __restrict__ out)
{
  __shared__ float tbuf[8][16][18];   // per-wave transpose scratch (9 KB)

  const int tid   = threadIdx.x;
  const int lane  = tid & 31;
  const int wave  = tid >> 5;
  const int m     = lane & 15;
  const int h     = lane >> 4;
  const int batch = blockIdx.y;
  const int qrow0 = blockIdx.x * 128 + wave * 16;
  const size_t base = (size_t)batch * SS;

  // A fragments for scores GEMM: A = qq tile [16 x 8], reused all key tiles.
  const float* qrow = qq + (base + qrow0 + m) * NWQ;
  v2f a0, a1;
  a0.x = qrow[2 * h];     a0.y = qrow[2 * h + 1];
  a1.x = qrow[4 + 2 * h]; a1.y = qrow[4 + 2 * h + 1];

  float Mx[8], Ls[8];
  v8f acc = {};
  #pragma unroll
  for (int j = 0; j < 8; ++j) { Mx[j] = -1e30f; Ls[j] = 0.0f; }

  const float scale = 0.35355339059327373f;  // 1/sqrt(NW)

  for (int kt = 0; kt < SS / 16; ++kt) {
    // B fragments: B[k][n] = kq[kt*16+n][k]; lane = key column n.
    const float* krow = kq + (base + kt * 16 + m) * NWQ;
    v2f b0, b1;
    b0.x = krow[2 * h];     b0.y = krow[2 * h + 1];
    b1.x = krow[4 + 2 * h]; b1.y = krow[4 + 2 * h + 1];

    v8f s = {};
    s = wmma_f32_4(a0, b0, s);
    s = wmma_f32_4(a1, b1, s);

    // Online softmax per query row (rows live across 16-lane halves).
    #pragma unroll
    for (int j = 0; j < 8; ++j) {
      float sv = s[j] * scale;
      float mx = sv;
      mx = fmaxf(mx, __shfl_xor(mx, 1));
      mx = fmaxf(mx, __shfl_xor(mx, 2));
      mx = fmaxf(mx, __shfl_xor(mx, 4));
      mx = fmaxf(mx, __shfl_xor(mx, 8));
      const float Mn   = fmaxf(Mx[j], mx);
      const float corr = __expf(Mx[j] - Mn);
      const float p    = __expf(sv - Mn);
      float ps = p;
      ps += __shfl_xor(ps, 1);
      ps += __shfl_xor(ps, 2);
      ps += __shfl_xor(ps, 4);
      ps += __shfl_xor(ps, 8);
      Ls[j]  = Ls[j] * corr + ps;
      acc[j] *= corr;
      Mx[j]  = Mn;
      tbuf[wave][j + 8 * h][m] = p;     // D layout -> LDS [row][keycol]
    }
    asm volatile("s_wait_dscnt 0" ::: "memory");

    // P @ V: 4 WMMAs of K=4 over the 16 keys in this tile.
    const float* vbase = vq + (base + kt * 16) * NWQ;
    #pragma unroll
    for (int q = 0; q < 4; ++q) {
      v2f pa, vb;
      pa.x = tbuf[wave][m][q * 4 + 2 * h];       // A layout: lane = query row
      pa.y = tbuf[wave][m][q * 4 + 2 * h + 1];
      float t0 = vbase[(size_t)(q * 4 + 2 * h)     * NWQ + (m & 7)];
      float t1 = vbase[(size_t)(q * 4 + 2 * h + 1) * NWQ + (m & 7)];
      vb.x = (m < 8) ? t0 : 0.0f;                // N columns 8..15 padded
      vb.y = (m < 8) ? t1 : 0.0f;
      acc = wmma_f32_4(pa, vb, acc);
    }
  }

  // Normalize and transpose O into A layout for the output projection.
  #pragma unroll
  for (int j = 0; j < 8; ++j)
    tbuf[wave][j + 8 * h][m] = acc[j] / Ls[j];
  asm volatile("s_wait_dscnt 0" ::: "memory");

  v2f fa0, fa1;
  fa0.x = tbuf[wave][m][2 * h];     fa0.y = tbuf[wave][m][2 * h + 1];
  fa1.x = tbuf[wave][m][4 + 2 * h]; fa1.y = tbuf[wave][m][4 + 2 * h + 1];

  // out[row][:] = O @ wc^T + bc ; B[k][e] = wc[e][k] (wc rows contiguous).
  float* orow = out + (base + qrow0) * EE;
  for (int et = 0; et < EE / 16; ++et) {
    const int e = et * 16 + m;
    const float* wcr = wc + (size_t)e * NWQ;
    v2f b0, b1;
    b0.x = wcr[2 * h];     b0.y = wcr[2 * h + 1];
    b1.x = wcr[4 + 2 * h]; b1.y = wcr[4 + 2 * h + 1];
    v8f d = {};
    d = wmma_f32_4(fa0, b0, d);
    d = wmma_f32_4(fa1, b1, d);
    const float bias = bc[e];
    #pragma unroll
    for (int j = 0; j < 8; ++j)
      orow[(size_t)(j + 8 * h) * EE + e] = d[j] + bias;  // 64B-coalesced per half
  }
}

// -------------------------------------------------------------------------
extern "C" void kernel_launch(void* const* d_in, const int* in_sizes, int n_in,
                              void* d_out, int out_size, void* d_ws, size_t ws_size,
                              hipStream_t stream) {
  const float* x     = (const float*)d_in[0];
  const float* wq    = (const float*)d_in[1];
  const float* bq    = (const float*)d_in[2];
  const float* wk    = (const float*)d_in[3];
  const float* bk    = (const float*)d_in[4];
  const float* wv    = (const float*)d_in[5];
  const float* bv    = (const float*)d_in[6];
  const float* wc    = (const float*)d_in[7];
  const float* bc    = (const float*)d_in[8];
  const float* theta = (const float*)d_in[9];

  float* qq = (float*)d_ws;                       // [B*S, 8]
  float* kq = qq + (size_t)BB * SS * NWQ;         // [B*S, 8]
  float* vq = kq + (size_t)BB * SS * NWQ;         // [B*S, 8]

  qkv_proj_kernel<<<dim3(BB * SS / 64), dim3(256), 0, stream>>>(
      x, wq, bq, wk, bk, wv, bv, theta, qq, kq, vq);

  attn_out_kernel<<<dim3(SS / 128, BB), dim3(256), 0, stream>>>(
      qq, kq, vq, wc, bc, (float*)d_out);
}